// RMSD_20126216749374
// MI455X (gfx1250) — compile-verified
//
#include <hip/hip_runtime.h>
#include <math.h>

// CDNA5 (gfx1250) wave32 WMMA vector types
typedef __attribute__((ext_vector_type(2))) float v2f;  // 16x4 f32 A/B fragment: 2 VGPRs
typedef __attribute__((ext_vector_type(8))) float v8f;  // 16x16 f32 C/D: 8 VGPRs

#define KB_NPTS   512
#define KB_NBATCH 8192

// ---------------------------------------------------------------------------
// Kernel 1: per-batch 7x7 Gram matrix via V_WMMA_F32_16X16X4_F32, then
// closed-form Kabsch loss for that batch.
//
// Augmented vector per point: v = (x0,x1,x2, y0,y1,y2, 1, 0,...,0)  (16 comps)
// G = sum_n v v^T contains every statistic we need:
//   G[i][3+j] = (x^T y)[i][j],  G[i][6]=sum x_i, G[3+j][6]=sum y_j,
//   diag 0..2 -> sum||x||^2,    diag 3..5 -> sum||y||^2, G[6][6]=N.
//
// WMMA f32 16x16x4 fragment layout (ISA 7.12.2, 32-bit A 16x4):
//   VGPR0: lanes 0-15 -> K=0, lanes 16-31 -> K=2
//   VGPR1: lanes 0-15 -> K=1, lanes 16-31 -> K=3
// B (4x16) is the lane-transpose, so one fragment serves as both A and B.
//
// Fragment construction is BRANCHLESS: every lane does an unconditional
// global_load_b32 from a valid per-lane address (dummy y[k*3] for lanes whose
// component is the constant 1 / zero padding), then one v_cndmask swaps in
// the constant. No exec-mask churn, loop fully unrolls to loads at immediate
// offsets + wmma.
// ---------------------------------------------------------------------------
__global__ __launch_bounds__(256) void kabsch_gram_wmma(
    const float* __restrict__ x, const float* __restrict__ y,
    float* __restrict__ per_batch)
{
  const int b   = blockIdx.x;
  const int tid = threadIdx.x;
  const int wv  = tid >> 5;   // wave 0..7
  const int ln  = tid & 31;
  const int hf  = ln >> 4;    // lane half selects K pair (0,1 or 2,3)
  const int cm  = ln & 15;    // matrix row/col index (component)

  const float* __restrict__ xb = x + (size_t)b * (KB_NPTS * 3);
  const float* __restrict__ yb = y + (size_t)b * (KB_NPTS * 3);

  // Per-lane source: comps 0-2 read x, comps 3-5 read y, comps 6-15 read a
  // harmless dummy (y[k*3]) that a select replaces with 1.0 / 0.0.
  const bool  use_mem  = (cm < 6);
  const float cst      = (cm == 6) ? 1.0f : 0.0f;
  const int   cmc      = (cm < 3) ? cm : ((cm < 6) ? (cm - 3) : 0);
  const float* basep   = (cm < 3) ? xb : yb;

  // first point this lane touches: k = pbase + 2*hf  (VGPR0 of step 0)
  const int pbase = wv * (KB_NPTS / 8);          // 64 points per wave
  const float* __restrict__ lanep = basep + ((pbase + 2 * hf) * 3 + cmc);

  v8f acc = {};

  #pragma unroll
  for (int s = 0; s < (KB_NPTS / 8) / 4; ++s) {  // 16 WMMA steps, K=4 points each
    // step s covers points k0 = pbase + 4s + 2hf (VGPR0) and k0+1 (VGPR1)
    const float m0 = lanep[s * 12];        // element offset 12 per step (4 pts * 3)
    const float m1 = lanep[s * 12 + 3];    // next point: +3 elements
    v2f frag;
    frag.x = use_mem ? m0 : cst;
    frag.y = use_mem ? m1 : cst;
    // D = frag * frag^T + acc  -> accumulates the 7x7 Gram in the corner
    acc = __builtin_amdgcn_wmma_f32_16x16x4_f32(
        false, frag, false, frag, (short)0, acc, false, false);
  }

  // --- combine the 8 per-wave partial Grams in a FIXED order (deterministic) ---
  // C/D layout: VGPR r, lanes 0-15 hold row M=r, col N=lane. Rows 0..6 of the
  // Gram therefore live in acc[0..6] of lanes 0..6 (half 0).
  __shared__ float sg[8][49];
  __shared__ float g[49];
  if (hf == 0 && cm < 7) {
    #pragma unroll
    for (int r = 0; r < 7; ++r) sg[wv][r * 7 + cm] = acc[r];
  }
  __syncthreads();
  if (tid < 49) {
    float t = 0.0f;
    #pragma unroll
    for (int w = 0; w < 8; ++w) t += sg[w][tid];
    g[tid] = t;
  }
  __syncthreads();

  // --- per-batch finalization: centered covariance -> singular values -> loss ---
  if (tid == 0) {
    const double Ninv = 1.0 / (double)KB_NPTS;
    double sx[3], sy[3];
    #pragma unroll
    for (int i = 0; i < 3; ++i) { sx[i] = g[i * 7 + 6]; sy[i] = g[(3 + i) * 7 + 6]; }

    double sqx = (double)g[0] + (double)g[8] + (double)g[16]
               - (sx[0]*sx[0] + sx[1]*sx[1] + sx[2]*sx[2]) * Ninv;
    double sqy = (double)g[24] + (double)g[32] + (double)g[40]
               - (sy[0]*sy[0] + sy[1]*sy[1] + sy[2]*sy[2]) * Ninv;

    double C[3][3];
    #pragma unroll
    for (int i = 0; i < 3; ++i)
      #pragma unroll
      for (int j = 0; j < 3; ++j)
        C[i][j] = (double)g[i * 7 + 3 + j] - sx[i] * sy[j] * Ninv;

    const double det =
        C[0][0]*(C[1][1]*C[2][2] - C[1][2]*C[2][1])
      - C[0][1]*(C[1][0]*C[2][2] - C[1][2]*C[2][0])
      + C[0][2]*(C[1][0]*C[2][1] - C[1][1]*C[2][0]);

    // K = C^T C (symmetric 3x3); its eigenvalues are sigma_i^2
    double a00=0, a11=0, a22=0, a01=0, a02=0, a12=0;
    #pragma unroll
    for (int k = 0; k < 3; ++k) {
      a00 += C[k][0]*C[k][0]; a11 += C[k][1]*C[k][1]; a22 += C[k][2]*C[k][2];
      a01 += C[k][0]*C[k][1]; a02 += C[k][0]*C[k][2]; a12 += C[k][1]*C[k][2];
    }

    // trigonometric closed-form eigenvalues of a symmetric 3x3
    const double q  = (a00 + a11 + a22) * (1.0 / 3.0);
    const double p1 = a01*a01 + a02*a02 + a12*a12;
    const double b0 = a00 - q, b1 = a11 - q, b2 = a22 - q;
    const double p2 = b0*b0 + b1*b1 + b2*b2 + 2.0 * p1;
    const double p  = sqrt(fmax(p2, 0.0) * (1.0 / 6.0));
    double l1, l2, l3;
    if (p < 1e-30) {
      l1 = l2 = l3 = q;
    } else {
      const double ip  = 1.0 / p;
      const double d00 = b0*ip, d11 = b1*ip, d22 = b2*ip;
      const double d01 = a01*ip, d02 = a02*ip, d12 = a12*ip;
      double r = 0.5 * (d00*(d11*d22 - d12*d12)
                      - d01*(d01*d22 - d12*d02)
                      + d02*(d01*d12 - d11*d02));
      r = fmin(1.0, fmax(-1.0, r));
      const double phi = acos(r) * (1.0 / 3.0);
      l1 = q + 2.0 * p * cos(phi);
      l3 = q + 2.0 * p * cos(phi + 2.0943951023931953);  // + 2*pi/3 -> smallest
      l2 = 3.0 * q - l1 - l3;
    }
    const double s1 = sqrt(fmax(l1, 0.0));
    const double s2 = sqrt(fmax(l2, 0.0));
    const double s3 = sqrt(fmax(l3, 0.0));  // smallest singular value
    const double sgn = (det < 0.0) ? -1.0 : 1.0;

    per_batch[b] = (float)(sqx + sqy - 2.0 * (s1 + s2 + sgn * s3));
  }
}

// ---------------------------------------------------------------------------
// Kernel 2: deterministic tree reduction of the 8192 per-batch losses.
// ---------------------------------------------------------------------------
__global__ __launch_bounds__(256) void kabsch_final_reduce(
    const float* __restrict__ per_batch, float* __restrict__ out)
{
  __shared__ double sm[256];
  double acc = 0.0;
  for (int i = threadIdx.x; i < KB_NBATCH; i += 256) acc += (double)per_batch[i];
  sm[threadIdx.x] = acc;
  __syncthreads();
  for (int off = 128; off > 0; off >>= 1) {
    if ((int)threadIdx.x < off) sm[threadIdx.x] += sm[threadIdx.x + off];
    __syncthreads();
  }
  if (threadIdx.x == 0)
    out[0] = (float)(sm[0] / ((double)KB_NBATCH * (double)KB_NPTS * 3.0));
}

extern "C" void kernel_launch(void* const* d_in, const int* in_sizes, int n_in,
                              void* d_out, int out_size, void* d_ws, size_t ws_size,
                              hipStream_t stream) {
  const float* inp = (const float*)d_in[0];   // (8192, 512, 3) f32
  const float* tgt = (const float*)d_in[1];   // (8192, 512, 3) f32
  float* out = (float*)d_out;                 // scalar f32
  float* ws  = (float*)d_ws;                  // >= 8192 floats of scratch

  kabsch_gram_wmma<<<KB_NBATCH, 256, 0, stream>>>(inp, tgt, ws);
  kabsch_final_reduce<<<1, 256, 0, stream>>>(ws, out);
}